// AttentionPool_77687368450351
// MI455X (gfx1250) — compile-verified
//
#include <hip/hip_runtime.h>
#include <math.h>

typedef __attribute__((ext_vector_type(2))) float v2f;
typedef __attribute__((ext_vector_type(8))) float v8f;

#define B_   16
#define T_   4096
#define D_   1024
#define TPW  2                         // 16-token tiles per wave
#define TOK_PER_WAVE (16 * TPW)        // 32 tokens / wave
#define CHUNKS_PER_B (T_ / TOK_PER_WAVE)   // 128 partials per batch
#define NWAVES (B_ * CHUNKS_PER_B)         // 2048 waves
#define WAVES_PER_BLOCK 8
#define NBLOCKS (NWAVES / WAVES_PER_BLOCK) // 256 blocks

__device__ __forceinline__ float rl_f32(float v, int lane) {
  return __int_as_float(__builtin_amdgcn_readlane(__float_as_int(v), lane));
}

// Kernel 1: fused scores (WMMA f32 16x16x4) + online softmax + partial pooled acc.
__global__ void attnpool_partial(const float* __restrict__ hs,
                                 const int*   __restrict__ mask,
                                 const float* __restrict__ ctx,
                                 float* __restrict__ ws_acc,
                                 float* __restrict__ ws_m,
                                 float* __restrict__ ws_s) {
  const int lane  = threadIdx.x & 31;
  const int gwave = blockIdx.x * WAVES_PER_BLOCK + (threadIdx.x >> 5);
  const int b     = gwave / CHUNKS_PER_B;
  const int chunk = gwave % CHUNKS_PER_B;
  const int t0    = chunk * TOK_PER_WAVE;

  // Each lane owns 32 d-values: d = j*128 + lane*4 .. +3  (j = 0..7)
  float4 acc[8];
  #pragma unroll
  for (int j = 0; j < 8; ++j) acc[j] = make_float4(0.f, 0.f, 0.f, 0.f);
  float m_run = -INFINITY;
  float s_run = 0.f;

  const int ksel = 2 * (lane >> 4);   // A/B fragment K-offset select (lanes 16-31 -> K+2)

  for (int tile = 0; tile < TPW; ++tile) {
    const int tt = t0 + tile * 16;
    const float* rowbase = hs + ((size_t)b * T_ + tt + (lane & 15)) * D_;

    // --- scores for 16 tokens via V_WMMA_F32_16X16X4_F32 (exact fp32) ---
    v8f c = {0.f, 0.f, 0.f, 0.f, 0.f, 0.f, 0.f, 0.f};
    #pragma unroll 8
    for (int k0 = 0; k0 < D_; k0 += 4) {
      const int ko = k0 + ksel;
      v2f a  = *(const v2f*)(rowbase + ko);   // A: 16 tokens x 4 K
      v2f bb = *(const v2f*)(ctx + ko);       // B: ctx replicated across all 16 columns
      c = __builtin_amdgcn_wmma_f32_16x16x4_f32(false, a, false, bb,
                                                (short)0, c, false, false);
    }

    // D[m][n] identical over n; token m score: VGPR m (lanes 0-15) / m-8 (lanes 16-31)
    float sc[16];
    #pragma unroll
    for (int v = 0; v < 8; ++v) {
      sc[v]     = rl_f32(c[v], 0);
      sc[v + 8] = rl_f32(c[v], 16);
    }
    #pragma unroll
    for (int i = 0; i < 16; ++i)
      if (mask[(size_t)b * T_ + tt + i] == 0) sc[i] = -INFINITY;

    float mt = sc[0];
    #pragma unroll
    for (int i = 1; i < 16; ++i) mt = fmaxf(mt, sc[i]);

    if (mt != -INFINITY) {               // wave-uniform branch
      const float newm  = fmaxf(m_run, mt);
      const float scale = expf(m_run - newm);   // 0 when m_run == -inf
      s_run *= scale;
      #pragma unroll
      for (int j = 0; j < 8; ++j) {
        acc[j].x *= scale; acc[j].y *= scale; acc[j].z *= scale; acc[j].w *= scale;
      }
      float w[16];
      #pragma unroll
      for (int i = 0; i < 16; ++i) { w[i] = expf(sc[i] - newm); s_run += w[i]; }

      // --- weighted accumulation; tile (64 KB) just streamed -> L2 hit ---
      #pragma unroll 2
      for (int mm = 0; mm < 16; ++mm) {
        const float* row = hs + ((size_t)b * T_ + tt + mm) * D_;
        #pragma unroll
        for (int j = 0; j < 8; ++j) {
          float4 hv = *(const float4*)(row + j * 128 + lane * 4);
          acc[j].x += w[mm] * hv.x; acc[j].y += w[mm] * hv.y;
          acc[j].z += w[mm] * hv.z; acc[j].w += w[mm] * hv.w;
        }
      }
      m_run = newm;
    }
  }

  float* outp = ws_acc + (size_t)gwave * D_;
  #pragma unroll
  for (int j = 0; j < 8; ++j)
    *(float4*)(outp + j * 128 + lane * 4) = acc[j];
  if (lane == 0) { ws_m[gwave] = m_run; ws_s[gwave] = s_run; }
}

// Kernel 2: merge 128 partials per batch (tiny: ~525 KB/block).
__global__ void attnpool_combine(const float* __restrict__ ws_acc,
                                 const float* __restrict__ ws_m,
                                 const float* __restrict__ ws_s,
                                 float* __restrict__ out) {
  const int b   = blockIdx.x;
  const int tid = threadIdx.x;
  __shared__ float s_m[CHUNKS_PER_B];
  __shared__ float s_s[CHUNKS_PER_B];
  __shared__ float s_scale[CHUNKS_PER_B];
  const int pbase = b * CHUNKS_PER_B;

  if (tid < CHUNKS_PER_B) { s_m[tid] = ws_m[pbase + tid]; s_s[tid] = ws_s[pbase + tid]; }
  __syncthreads();

  float M = -INFINITY;
  for (int p = 0; p < CHUNKS_PER_B; ++p) M = fmaxf(M, s_m[p]);
  float S = 0.f;
  for (int p = 0; p < CHUNKS_PER_B; ++p) S += expf(s_m[p] - M) * s_s[p];
  if (tid < CHUNKS_PER_B) s_scale[tid] = expf(s_m[tid] - M);
  __syncthreads();

  const float inv = 1.f / S;
  for (int d = tid; d < D_; d += blockDim.x) {
    float sum = 0.f;
    for (int p = 0; p < CHUNKS_PER_B; ++p)
      sum += s_scale[p] * ws_acc[(size_t)(pbase + p) * D_ + d];
    out[(size_t)b * D_ + d] = sum * inv;
  }
}

extern "C" void kernel_launch(void* const* d_in, const int* in_sizes, int n_in,
                              void* d_out, int out_size, void* d_ws, size_t ws_size,
                              hipStream_t stream) {
  (void)in_sizes; (void)n_in; (void)out_size; (void)ws_size;
  const float* hs  = (const float*)d_in[0];   // (16,4096,1024) f32
  const int*   mk  = (const int*)d_in[1];     // (16,4096) i32
  const float* ctx = (const float*)d_in[2];   // (1024,) f32
  float* out = (float*)d_out;                 // (16,1024) f32

  float* ws_acc = (float*)d_ws;                       // 2048*1024 f32 (8 MB)
  float* ws_m   = ws_acc + (size_t)NWAVES * D_;       // 2048 f32
  float* ws_s   = ws_m + NWAVES;                      // 2048 f32

  attnpool_partial<<<NBLOCKS, 256, 0, stream>>>(hs, mk, ctx, ws_acc, ws_m, ws_s);
  attnpool_combine<<<B_, 256, 0, stream>>>(ws_acc, ws_m, ws_s, out);
}